// MLP_64991445123151
// MI455X (gfx1250) — compile-verified
//
#include <hip/hip_runtime.h>
#include <math.h>

// ---------------------------------------------------------------------------
// CDNA5 (gfx1250): fp32 WMMA (V_WMMA_F32_16X16X4_F32) for all matmuls,
// eigh replaced by coupled Newton-Schulz, and async global->LDS double
// buffering (GLOBAL_LOAD_ASYNC_TO_LDS_B128 + s_wait_asynccnt) for the
// SPD tile stream in the barycenter.
// ---------------------------------------------------------------------------

typedef float v2f __attribute__((ext_vector_type(2)));
typedef float v8f __attribute__((ext_vector_type(8)));

#define DEVINL __device__ __forceinline__

// Wave-level ordering fence for LDS cross-lane communication.
// LDS ops are in-order within a wave (hardware); this pins compile-time order.
DEVINL void wave_sync() {
  __builtin_amdgcn_wave_barrier();
  asm volatile("" ::: "memory");
}

// Async copy of one 16x16 f32 matrix (1KB) global -> LDS.
// Each lane moves 16B per instruction; INST_OFFSET is added to BOTH the
// global and LDS addresses, so offset:512 covers the second half.
// Tracked by ASYNCcnt (increments by 2 per call).
DEVINL void async_copy_mat(const float* __restrict__ gsrc, float* lds_dst) {
  const int lane = threadIdx.x & 31;
  const unsigned long long gaddr =
      (unsigned long long)(const char*)gsrc + (unsigned)(lane * 16);
  const unsigned ldsoff =
      (unsigned)(unsigned long long)lds_dst + (unsigned)(lane * 16);
  asm volatile("global_load_async_to_lds_b128 %0, %1, off"
               :: "v"(ldsoff), "v"(gaddr) : "memory");
  asm volatile("global_load_async_to_lds_b128 %0, %1, off offset:512"
               :: "v"(ldsoff), "v"(gaddr) : "memory");
}

// ---- 16x16 f32 matmul on one wave32: C(16x16) = P(16x16) @ Q(16x16) -------
// A-fragment (16x4 f32, 2 VGPRs): lanes 0-15 hold M=lane, K = kk*4 + {0,1};
// lanes 16-31 hold M=lane-16, K = kk*4 + {2,3}.
// B-fragment (4x16 f32, 2 VGPRs): lanes 0-15 hold N=lane, K = kk*4 + {0,1};
// lanes 16-31 hold N=lane-16, K = kk*4 + {2,3}.
DEVINL v8f mat16_mul(const float* P, const float* Q) {
  const int lane = threadIdx.x & 31;
  const int l = lane & 15;
  const int hoff = (lane >> 4) << 1;  // 0 or 2
  v8f acc = {};
#pragma unroll
  for (int kk = 0; kk < 4; ++kk) {
    const int ka = kk * 4 + hoff;
    v2f a, b;
    a.x = P[l * 16 + ka];
    a.y = P[l * 16 + ka + 1];
    b.x = Q[ka * 16 + l];
    b.y = Q[(ka + 1) * 16 + l];
    acc = __builtin_amdgcn_wmma_f32_16x16x4_f32(false, a, false, b,
                                                (short)0, acc, false, false);
  }
  return acc;
}

// C/D layout (8 VGPRs): vgpr r = row r (lanes 0-15) / row r+8 (lanes 16-31),
// col = lane & 15.  Store scaled accumulator to a row-major 16x16 buffer.
DEVINL void frag_store(float* D, v8f acc, float scale) {
  const int lane = threadIdx.x & 31;
  const int l = lane & 15;
  const int rb = (lane >> 4) << 3;  // 0 or 8
#pragma unroll
  for (int r = 0; r < 8; ++r) D[(rb + r) * 16 + l] = acc[r] * scale;
}

DEVINL float mat16_trace(const float* M) {
  const int lane = threadIdx.x & 31;
  float p = (lane < 16) ? M[lane * 17] : 0.0f;
#pragma unroll
  for (int m = 16; m >= 1; m >>= 1) p += __shfl_xor(p, m, 32);
  return p;
}

// Coupled Newton-Schulz: on entry Y holds SPD matrix M (row-major, LDS).
// On exit: Y ~= sqrtm(M)/s,  Z ~= invsqrtm(M)*s,  with s = sqrt(trace(M)).
// Returns s.  Pure 16x16 WMMA matmuls.
DEVINL float ns_pair(float* Y, float* Z, float* T, int iters) {
  const int lane = threadIdx.x & 31;
  const int t8 = lane * 8;
  float c = mat16_trace(Y);
  c = fmaxf(c, 1e-12f);
  const float inv_c = 1.0f / c;
#pragma unroll
  for (int j = 0; j < 8; ++j) {
    const int idx = t8 + j;
    Y[idx] *= inv_c;                                    // Y0 = M / c
    Z[idx] = ((idx >> 4) == (idx & 15)) ? 1.0f : 0.0f;  // Z0 = I
  }
  wave_sync();
  for (int t = 0; t < iters; ++t) {
    v8f tac = mat16_mul(Z, Y);  // T = Z @ Y
    {                           // E = 3I - T  (stored into T, C/D layout)
      const int l = lane & 15;
      const int rb = (lane >> 4) << 3;
#pragma unroll
      for (int r = 0; r < 8; ++r) {
        const int row = rb + r;
        T[row * 16 + l] = ((row == l) ? 3.0f : 0.0f) - tac[r];
      }
    }
    wave_sync();
    v8f yac = mat16_mul(Y, T);  // Y' = 0.5 * Y @ E
    v8f zac = mat16_mul(T, Z);  // Z' = 0.5 * E @ Z
    wave_sync();
    frag_store(Y, yac, 0.5f);
    frag_store(Z, zac, 0.5f);
    wave_sync();
  }
  return sqrtf(c);
}

// ---------------------------------------------------------------------------
// MLP GEMM: Z = A(MxK) @ W(KxN) + bias.  One wave per 16x16 output tile,
// fragments loaded straight from global in WMMA layouts (weights are L2-hot).
// ---------------------------------------------------------------------------
__global__ __launch_bounds__(128) void gemm_bias_kernel(
    const float* __restrict__ A, const float* __restrict__ W,
    const float* __restrict__ bias, float* __restrict__ Z,
    int K, int N) {
  const int wave = threadIdx.x >> 5;
  const int lane = threadIdx.x & 31;
  const int l = lane & 15;
  const int hoff = (lane >> 4) << 1;  // 0 or 2
  const int tm = blockIdx.x * 16;
  const int tn = (blockIdx.y * 4 + wave) * 16;
  if (tn >= N) return;  // wave-uniform
  const int row = tm + l;
  const int col = tn + l;
  v8f acc = {};
  for (int k0 = 0; k0 < K; k0 += 4) {
    const int ka = k0 + hoff;
    v2f a, b;
    a.x = A[row * K + ka];
    a.y = A[row * K + ka + 1];
    b.x = W[ka * N + col];
    b.y = W[(ka + 1) * N + col];
    acc = __builtin_amdgcn_wmma_f32_16x16x4_f32(false, a, false, b,
                                                (short)0, acc, false, false);
  }
  const float bv = bias[col];
  const int rb = (lane >> 4) << 3;
#pragma unroll
  for (int r = 0; r < 8; ++r) Z[(tm + rb + r) * N + col] = acc[r] + bv;
}

// ---------------------------------------------------------------------------
// norm_e2m over axis=-2 (the 256-row batch), per column.
// mode 0: relu(norm)      mode 1: Hin + relu(norm)      mode 2: norm only
// ---------------------------------------------------------------------------
__global__ void norm_kernel(const float* __restrict__ Z,
                            const float* __restrict__ alpha,
                            const float* __restrict__ bias,
                            const float* __restrict__ Hin,
                            float* __restrict__ Hout, int Ncols, int mode) {
  const int c = blockIdx.x * blockDim.x + threadIdx.x;
  if (c >= Ncols) return;
  float sum = 0.f, sq = 0.f;
  for (int r = 0; r < 256; ++r) {
    const float v = Z[r * Ncols + c];
    sum += v;
    sq += v * v;
  }
  const float mean = sum * (1.0f / 256.0f);
  const float var = fmaxf(sq * (1.0f / 256.0f) - mean * mean, 0.0f);
  const float istd = 1.0f / (sqrtf(var) + 1e-6f);
  const float al = alpha[c], bi = bias[c];
  for (int r = 0; r < 256; ++r) {
    float v = al * (Z[r * Ncols + c] - mean) * istd + bi;
    if (mode == 0) v = fmaxf(v, 0.0f);
    else if (mode == 1) v = Hin[r * Ncols + c] + fmaxf(v, 0.0f);
    Hout[r * Ncols + c] = v;
  }
}

__global__ __launch_bounds__(128) void softmax_kernel(
    const float* __restrict__ L, float* __restrict__ Wsm) {
  __shared__ float red[128];
  const int r = blockIdx.x, c = threadIdx.x;
  const float v = L[r * 128 + c];
  red[c] = v;
  __syncthreads();
  for (int s = 64; s > 0; s >>= 1) {
    if (c < s) red[c] = fmaxf(red[c], red[c + s]);
    __syncthreads();
  }
  const float mx = red[0];
  __syncthreads();
  const float e = __expf(v - mx);
  red[c] = e;
  __syncthreads();
  for (int s = 64; s > 0; s >>= 1) {
    if (c < s) red[c] += red[c + s];
    __syncthreads();
  }
  Wsm[r * 128 + c] = e / red[0];
}

// ---------------------------------------------------------------------------
// Bures-Wasserstein barycenter: one block (8 wave32s) per batch element.
// Eigh replaced by Newton-Schulz; 16x16 fp32 WMMA throughout; the A_i SPD
// tile stream is double-buffered in LDS via async global->LDS copies.
// ---------------------------------------------------------------------------
#define NW 8
#define NS_IT 8
#define OUTER_IT 10

__global__ __launch_bounds__(256) void bary_kernel(
    const float* __restrict__ Atr,   // y_train: 128 x (16*16), row-major SPD
    const float* __restrict__ Wgt,   // softmax weights: 256 x 128
    float* __restrict__ Out) {       // 256 x 256
  __shared__ float sS[256], sSq[256], sIs[256], sWs[256];
  __shared__ float scr[NW][3][256];
  __shared__ float sAbuf[NW][2][256];  // per-wave double buffer for A_i tiles
  const int tid = threadIdx.x;
  const int wave = tid >> 5;
  const int lane = tid & 31;
  const int t8 = lane * 8;
  const int b = blockIdx.x;
  float* Y = scr[wave][0];
  float* Z = scr[wave][1];
  float* T = scr[wave][2];

  sS[tid] = ((tid >> 4) == (tid & 15)) ? 1.0f : 0.0f;  // S0 = I
  __syncthreads();

  for (int it = 0; it < OUTER_IT; ++it) {
    sWs[tid] = 0.0f;

    // Kick off the async prefetch of this wave's first A tile while the
    // (redundant, wave-uniform) NS on S runs.
    async_copy_mat(Atr + wave * 256, sAbuf[wave][0]);

    // sqrtm(S), invsqrtm(S): computed redundantly by every wave (uniform flow)
#pragma unroll
    for (int j = 0; j < 8; ++j) Y[t8 + j] = sS[t8 + j];
    wave_sync();
    const float s = ns_pair(Y, Z, T, NS_IT);
    const float is = 1.0f / s;
#pragma unroll
    for (int j = 0; j < 8; ++j) {
      sSq[t8 + j] = Y[t8 + j] * s;   // sqrt(S)
      sIs[t8 + j] = Z[t8 + j] * is;  // invsqrt(S)
    }
    __syncthreads();

    // sandwich_i = sqrtS @ A_i @ sqrtS; sqrtm via NS; weighted accumulate.
    // A tiles stream through the per-wave LDS double buffer (ASYNCcnt).
    float wacc[8] = {0, 0, 0, 0, 0, 0, 0, 0};
    int ibuf = 0;
    for (int i = wave; i < 128; i += NW) {
      float* Acur = sAbuf[wave][ibuf];
      float* Anxt = sAbuf[wave][ibuf ^ 1];
      const int inext = i + NW;
      if (inext < 128) {  // wave-uniform branch
        async_copy_mat(Atr + inext * 256, Anxt);
        // 2 copies in flight for the next tile; wait until current tile's
        // 2 copies have retired (async loads complete in order).
        asm volatile("s_wait_asynccnt 0x2" ::: "memory");
      } else {
        asm volatile("s_wait_asynccnt 0x0" ::: "memory");
      }
      wave_sync();
      v8f t1 = mat16_mul(sSq, Acur);  // T1 = sqrtS @ A_i
      frag_store(T, t1, 1.0f);
      wave_sync();
      v8f m = mat16_mul(T, sSq);      // M = T1 @ sqrtS
      frag_store(Y, m, 1.0f);
      wave_sync();
      const float si = ns_pair(Y, Z, T, NS_IT);  // Y ~= sqrtm(M)/si
      const float wi = Wgt[b * 128 + i] * si;
#pragma unroll
      for (int j = 0; j < 8; ++j) wacc[j] += wi * Y[t8 + j];
      wave_sync();
      ibuf ^= 1;
    }
#pragma unroll
    for (int j = 0; j < 8; ++j) atomicAdd(&sWs[t8 + j], wacc[j]);
    __syncthreads();

    // S <- invsqrtS @ (Ws @ Ws) @ invsqrtS   (redundant on all waves)
    v8f p = mat16_mul(sWs, sWs);
    frag_store(T, p, 1.0f);
    wave_sync();
    v8f q = mat16_mul(sIs, T);
    frag_store(Y, q, 1.0f);
    wave_sync();
    v8f r2 = mat16_mul(Y, sIs);
    frag_store(Z, r2, 1.0f);
    wave_sync();
    __syncthreads();  // everyone done reading sWs/sIs
#pragma unroll
    for (int j = 0; j < 8; ++j) sS[t8 + j] = Z[t8 + j];  // identical values
    __syncthreads();
  }
  Out[b * 256 + tid] = sS[tid];
}

// ---------------------------------------------------------------------------
extern "C" void kernel_launch(void* const* d_in, const int* in_sizes, int n_in,
                              void* d_out, int out_size, void* d_ws,
                              size_t ws_size, hipStream_t stream) {
  (void)in_sizes; (void)n_in; (void)out_size; (void)ws_size;
  const float* x         = (const float*)d_in[0];
  const float* y_train   = (const float*)d_in[1];
  const float* W1        = (const float*)d_in[2];
  const float* b1        = (const float*)d_in[3];
  const float* alpha1    = (const float*)d_in[4];
  const float* bias1     = (const float*)d_in[5];
  const float* Wh        = (const float*)d_in[6];
  const float* bh        = (const float*)d_in[7];
  const float* alpha_h   = (const float*)d_in[8];
  const float* bias_h    = (const float*)d_in[9];
  const float* Wout      = (const float*)d_in[10];
  const float* bout      = (const float*)d_in[11];
  const float* alpha_out = (const float*)d_in[12];
  const float* bias_out  = (const float*)d_in[13];
  float* out = (float*)d_out;

  const int B = 256, F = 512, H = 1024, N = 128;
  float* h   = (float*)d_ws;       // B*H
  float* z   = h + B * H;          // B*H
  float* wgt = z + B * H;          // B*N

  // h = relu(norm(x @ W1 + b1))
  gemm_bias_kernel<<<dim3(B / 16, H / 64), 128, 0, stream>>>(x, W1, b1, z, F, H);
  norm_kernel<<<dim3(H / 256), 256, 0, stream>>>(z, alpha1, bias1, nullptr, h, H, 0);

  // residual hidden layers
  for (int l = 0; l < 3; ++l) {
    gemm_bias_kernel<<<dim3(B / 16, H / 64), 128, 0, stream>>>(
        h, Wh + (size_t)l * H * H, bh + l * H, z, H, H);
    norm_kernel<<<dim3(H / 256), 256, 0, stream>>>(
        z, alpha_h + l * H, bias_h + l * H, h, h, H, 1);
  }

  // logits = norm(h @ Wout + bout); w = softmax(logits, axis=1)
  gemm_bias_kernel<<<dim3(B / 16, N / 64), 128, 0, stream>>>(h, Wout, bout, z, H, N);
  norm_kernel<<<dim3(1), 256, 0, stream>>>(z, alpha_out, bias_out, nullptr, wgt, N, 2);
  softmax_kernel<<<dim3(B), 128, 0, stream>>>(wgt, wgt);

  // Bures-Wasserstein barycenter
  bary_kernel<<<dim3(B), 256, 0, stream>>>(y_train, wgt, out);
}